// SelectTopk_3745211483118
// MI455X (gfx1250) — compile-verified
//
#include <hip/hip_runtime.h>
#include <math.h>
#include <stdint.h>

// MoE router: softmax -> top-8 -> renormalize, for T=262144 tokens x 128 experts.
// Lane-per-token branchless top-8 via v_med3_f32 insertion network.
// Rows staged global->LDS with gfx1250 async copies (ASYNCcnt).

#define EXPERTS          128
#define TOPK             8
#define ROW_BYTES        512                  // 128 floats
#define ROW_STRIDE_LDS   528                  // 512 + 16B pad (16B aligned, kills worst LDS conflicts)
#define TOKENS_PER_WAVE  32                   // one token per lane (wave32)
#define WAVES_PER_BLOCK  2
#define TOKENS_PER_BLOCK (WAVES_PER_BLOCK * TOKENS_PER_WAVE)   // 64
#define NEG_INF          (-3.402823466e+38f)

__device__ __forceinline__ float med3f(float a, float b, float c) {
#if __has_builtin(__builtin_amdgcn_fmed3f)
    return __builtin_amdgcn_fmed3f(a, b, c);
#else
    return fminf(fmaxf(a, b), c);   // valid because b <= c in our usage
#endif
}

// Insert x (global index J) into ascending top-8 (s0..s7 values, i0..i7 ids).
// s'_k = med3(x, s_k, s_{k+1}); index update driven by strict compares so
// equal values keep the earlier (lower-index) element ranked higher, matching
// jax.lax.top_k tie-breaking.
#define TOP8_INSERT(x, J)                                                    \
    do {                                                                     \
        const float _x = (x);                                                \
        const int   _j = (J);                                                \
        bool c0 = _x > s0, c1 = _x > s1, c2 = _x > s2, c3 = _x > s3;         \
        bool c4 = _x > s4, c5 = _x > s5, c6 = _x > s6, c7 = _x > s7;         \
        i0 = c0 ? (c1 ? i1 : _j) : i0;                                       \
        i1 = c1 ? (c2 ? i2 : _j) : i1;                                       \
        i2 = c2 ? (c3 ? i3 : _j) : i2;                                       \
        i3 = c3 ? (c4 ? i4 : _j) : i3;                                       \
        i4 = c4 ? (c5 ? i5 : _j) : i4;                                       \
        i5 = c5 ? (c6 ? i6 : _j) : i5;                                       \
        i6 = c6 ? (c7 ? i7 : _j) : i6;                                       \
        i7 = c7 ? _j : i7;                                                   \
        s0 = med3f(_x, s0, s1);                                              \
        s1 = med3f(_x, s1, s2);                                              \
        s2 = med3f(_x, s2, s3);                                              \
        s3 = med3f(_x, s3, s4);                                              \
        s4 = med3f(_x, s4, s5);                                              \
        s5 = med3f(_x, s5, s6);                                              \
        s6 = med3f(_x, s6, s7);                                              \
        s7 = fmaxf(s7, _x);                                                  \
    } while (0)

__global__ __launch_bounds__(TOKENS_PER_BLOCK)
void moe_topk_router(const float* __restrict__ logits,
                     int* __restrict__ out_ids,
                     float* __restrict__ out_wts,
                     int num_tokens) {
    __shared__ __align__(16) unsigned char smem[WAVES_PER_BLOCK * TOKENS_PER_WAVE * ROW_STRIDE_LDS];

    const int lane          = threadIdx.x & 31;
    const int waveId        = threadIdx.x >> 5;
    const int waveTokenBase = blockIdx.x * TOKENS_PER_BLOCK + waveId * TOKENS_PER_WAVE;
    const unsigned waveLds  = (unsigned)waveId * (TOKENS_PER_WAVE * ROW_STRIDE_LDS);

    // ---- Stage 1: async copy this wave's 32 rows, global -> LDS ------------
    // Row r: lanes cover 512B contiguously (coalesced); LDS rows padded to 528B.
    {
        const unsigned long long sbase = (unsigned long long)(uintptr_t)logits;
        const unsigned smemBase = (unsigned)(uintptr_t)&smem[0];
        #pragma unroll 4
        for (int r = 0; r < TOKENS_PER_WAVE; ++r) {
            int tok = waveTokenBase + r;
            if (tok >= num_tokens) tok = num_tokens - 1;      // clamp: no OOB reads
            unsigned vaddr = (unsigned)tok * ROW_BYTES + (unsigned)lane * 16u;
            unsigned ldsa  = smemBase + waveLds + (unsigned)r * ROW_STRIDE_LDS
                           + (unsigned)lane * 16u;
            asm volatile("global_load_async_to_lds_b128 %0, %1, %2"
                         :: "v"(ldsa), "v"(vaddr), "s"(sbase) : "memory");
        }
        asm volatile("s_wait_asynccnt 0" ::: "memory");
    }

    // ---- Stage 2: per-lane branchless top-8 over this lane's row -----------
    const int tok = waveTokenBase + lane;
    const float4* __restrict__ row =
        (const float4*)(smem + waveLds + (unsigned)lane * ROW_STRIDE_LDS);

    float s0 = NEG_INF, s1 = NEG_INF, s2 = NEG_INF, s3 = NEG_INF;
    float s4 = NEG_INF, s5 = NEG_INF, s6 = NEG_INF, s7 = NEG_INF;
    int   i0 = 0, i1 = 0, i2 = 0, i3 = 0, i4 = 0, i5 = 0, i6 = 0, i7 = 0;

    #pragma unroll 4
    for (int c = 0; c < EXPERTS / 4; ++c) {
        const float4 x4 = row[c];       // ds_load_b128
        const int jb = c * 4;
        TOP8_INSERT(x4.x, jb + 0);
        TOP8_INSERT(x4.y, jb + 1);
        TOP8_INSERT(x4.z, jb + 2);
        TOP8_INSERT(x4.w, jb + 3);
    }

    if (tok < num_tokens) {
        // Renormalized weights: softmax over the top-8 logits (full-row
        // denominator cancels in the has_moe_norm renormalization).
        const float m  = s7;            // max of the row
        const float e7 = 1.0f;
        const float e6 = __expf(s6 - m);
        const float e5 = __expf(s5 - m);
        const float e4 = __expf(s4 - m);
        const float e3 = __expf(s3 - m);
        const float e2 = __expf(s2 - m);
        const float e1 = __expf(s1 - m);
        const float e0 = __expf(s0 - m);
        const float denom = ((e7 + e6) + (e5 + e4)) + ((e3 + e2) + (e1 + e0));
        const float inv   = 1.0f / denom;

        // Outputs in descending-probability order (s7 is the largest).
        int4   idA = make_int4(i7, i6, i5, i4);
        int4   idB = make_int4(i3, i2, i1, i0);
        float4 wA  = make_float4(e7 * inv, e6 * inv, e5 * inv, e4 * inv);
        float4 wB  = make_float4(e3 * inv, e2 * inv, e1 * inv, e0 * inv);

        const size_t base = (size_t)tok * TOPK;
        *(int4*)(out_ids + base)           = idA;
        *(int4*)(out_ids + base + 4)       = idB;
        *(float4*)(out_wts + base)         = wA;
        *(float4*)(out_wts + base + 4)     = wB;
    }
}

extern "C" void kernel_launch(void* const* d_in, const int* in_sizes, int n_in,
                              void* d_out, int out_size, void* d_ws, size_t ws_size,
                              hipStream_t stream) {
    (void)n_in; (void)d_ws; (void)ws_size; (void)out_size;
    const float* logits = (const float*)d_in[0];
    const int T = in_sizes[0] / EXPERTS;

    // d_out layout (tuple concatenated flat): int32 ids [T,8] then f32 weights [T,8]
    int*   ids = (int*)d_out;
    float* wts = (float*)((char*)d_out + (size_t)T * TOPK * sizeof(int));

    const int grid = (T + TOKENS_PER_BLOCK - 1) / TOKENS_PER_BLOCK;
    moe_topk_router<<<grid, TOKENS_PER_BLOCK, 0, stream>>>(logits, ids, wts, T);
}